// InnerProductLayer_5970004542170
// MI455X (gfx1250) — compile-verified
//
#include <hip/hip_runtime.h>

typedef __attribute__((ext_vector_type(2))) float v2f;
typedef __attribute__((ext_vector_type(8))) float v8f;

#define NF 64
#define ND 32
#define NPAIR 2016   // 64*63/2
#define WAVES_PER_BLOCK 8

// One wave32 per sample. Gram via v_wmma_f32_16x16x4_f32:
//   A (16x4) of row-block i and B (4x16) of row-block j share the identical
//   per-lane register image for a Gram matrix, so each K-step needs only four
//   float2 loads feeding ten 16x16 accumulator tiles (i<=j).
__global__ __launch_bounds__(256) void gram_upper_wmma_kernel(
    const float* __restrict__ X, float* __restrict__ out, int nbatch) {
  const int lane = threadIdx.x & 31;
  const int wave = threadIdx.x >> 5;
  const int b = blockIdx.x * WAVES_PER_BLOCK + wave;
  if (b >= nbatch) return;  // wave-uniform: EXEC stays all-ones for WMMA

  const float* xb = X + (size_t)b * (NF * ND);
  const int m = lane & 15;             // row within 16-block (A) / col (B)
  const int khalf = (lane >> 4) << 1;  // lanes 0-15 -> K{0,1}; 16-31 -> K{2,3}

  v8f acc[10];
#pragma unroll
  for (int t = 0; t < 10; ++t)
    acc[t] = (v8f){0.f, 0.f, 0.f, 0.f, 0.f, 0.f, 0.f, 0.f};

#pragma unroll
  for (int kk = 0; kk < ND; kk += 4) {
    v2f a[4];
#pragma unroll
    for (int rb = 0; rb < 4; ++rb)
      a[rb] = *(const v2f*)(xb + (rb * 16 + m) * ND + kk + khalf);

    int t = 0;
#pragma unroll
    for (int i = 0; i < 4; ++i) {
#pragma unroll
      for (int j = i; j < 4; ++j) {
        // D = a[i] (16x4) x a[j]^T-image (4x16) + C
        acc[t] = __builtin_amdgcn_wmma_f32_16x16x4_f32(
            /*neg_a=*/false, a[i], /*neg_b=*/false, a[j],
            /*c_mod=*/(short)0, acc[t], /*reuse_a=*/false, /*reuse_b=*/false);
        ++t;
      }
    }
  }

  // Scatter strict upper triangle in combinations(range(64), 2) order.
  float* ob = out + (size_t)b * NPAIR;
  const int rowoff = (lane >> 4) << 3;  // C/D layout: lanes 16-31 hold M+8
  int t = 0;
#pragma unroll
  for (int i = 0; i < 4; ++i) {
#pragma unroll
    for (int j = i; j < 4; ++j) {
#pragma unroll
      for (int r = 0; r < 8; ++r) {
        const int gi = i * 16 + r + rowoff;
        const int gj = j * 16 + m;
        if (gi < gj) {
          const int idx = gi * 63 - (gi * (gi - 1)) / 2 + (gj - gi - 1);
          ob[idx] = acc[t][r];
        }
      }
      ++t;
    }
  }
}

extern "C" void kernel_launch(void* const* d_in, const int* in_sizes, int n_in,
                              void* d_out, int out_size, void* d_ws, size_t ws_size,
                              hipStream_t stream) {
  const float* X = (const float*)d_in[0];
  float* out = (float*)d_out;
  const int nbatch = in_sizes[0] / (NF * ND);  // 16384
  const int grid = (nbatch + WAVES_PER_BLOCK - 1) / WAVES_PER_BLOCK;
  gram_upper_wmma_kernel<<<grid, 32 * WAVES_PER_BLOCK, 0, stream>>>(X, out, nbatch);
}